// ResidualBlock_1288490189548
// MI455X (gfx1250) — compile-verified
//
#include <hip/hip_runtime.h>

typedef __attribute__((ext_vector_type(2))) float v2f;
typedef __attribute__((ext_vector_type(8))) float v8f;

#define BN_EPS 1e-5f

// ---------------- degree / norm kernels ----------------

__global__ void deg_kernel(const int* __restrict__ src, const int* __restrict__ dst,
                           int* __restrict__ degs, int* __restrict__ degd, int E) {
    int i = blockIdx.x * blockDim.x + threadIdx.x;
    if (i < E) {
        atomicAdd(&degs[src[i]], 1);
        atomicAdd(&degd[dst[i]], 1);
    }
}

// in place: int degree bits -> float rsqrt(deg) (0 if deg==0)
__global__ void deg2norm_kernel(float* __restrict__ p, int n) {
    int i = blockIdx.x * blockDim.x + threadIdx.x;
    if (i < n) {
        int d = reinterpret_cast<const int*>(p)[i];
        p[i] = (d > 0) ? rsqrtf((float)d) : 0.0f;
    }
}

// ---------------- edge scatter: m[dst] += feat[src] * ns[src] ----------------
// one wave32 per edge; each lane owns 4 contiguous columns (float4 gather).
__global__ void scatter_kernel(const float* __restrict__ feat, const float* __restrict__ ns,
                               const int* __restrict__ src, const int* __restrict__ dst,
                               float* __restrict__ m, int E) {
    int edge = (int)((blockIdx.x * blockDim.x + threadIdx.x) >> 5);
    int lane = threadIdx.x & 31;
    if (edge >= E) return;
    int s = src[edge];
    int d = dst[edge];
    float sc = ns[s];
    float4 v = reinterpret_cast<const float4*>(feat + (size_t)s * 128)[lane];
    float* mp = m + (size_t)d * 128 + lane * 4;
    atomicAdd(mp + 0, v.x * sc);
    atomicAdd(mp + 1, v.y * sc);
    atomicAdd(mp + 2, v.z * sc);
    atomicAdd(mp + 3, v.w * sc);
}

// -------- WMMA GEMM fused with BN stats:
//   out = (m @ W) * nd[:,None] + bias ;  stats[c] += sum_col ; stats[128+c] += sumsq_col
// block = 256 threads = 8 waves; block handles 16 rows x 128 cols.
// A tile (16x128) staged in LDS once per block; wave w computes cols [16w,16w+16).
// K walked in steps of 4 with v_wmma_f32_16x16x4_f32 (fp32 end-to-end).
// C-layout: lane r holds rows 0-7 of column (16w+r), lane r+16 holds rows 8-15,
// so per-column stats need only a lane^16 shuffle + one atomic per column/block.
__global__ void __launch_bounds__(256)
gemm_bias_stats_kernel(const float* __restrict__ m, const float* __restrict__ W,
                       const float* __restrict__ nd, const float* __restrict__ bias,
                       float* __restrict__ outb, float* __restrict__ stats) {
    __shared__ float As[16 * 128];
    int tid  = threadIdx.x;
    int wave = tid >> 5;
    int lane = tid & 31;
    int row0 = blockIdx.x * 16;

    // cooperative A-tile load: 2048 floats = 512 float4, 256 threads x 2
    {
        const float4* g  = reinterpret_cast<const float4*>(m + (size_t)row0 * 128);
        float4*       s4 = reinterpret_cast<float4*>(As);
        s4[tid]       = g[tid];
        s4[tid + 256] = g[tid + 256];
    }
    __syncthreads();

    int r   = lane & 15;          // A row in tile / B,C column in tile
    int kp  = (lane >> 4) << 1;   // K sub-offset: 0 (lanes 0-15) or 2 (lanes 16-31)
    int col = wave * 16 + r;      // global output column

    v8f c = {};
#pragma unroll
    for (int k0 = 0; k0 < 128; k0 += 4) {
        v2f a, b;
        a.x = As[r * 128 + k0 + kp];
        a.y = As[r * 128 + k0 + kp + 1];
        b.x = W[(size_t)(k0 + kp) * 128 + col];
        b.y = W[(size_t)(k0 + kp + 1) * 128 + col];
        c = __builtin_amdgcn_wmma_f32_16x16x4_f32(false, a, false, b, (short)0, c, false, false);
    }

    int rbase = row0 + ((lane >> 4) << 3);  // +8 for upper half-lanes
    float bcol = bias[col];
    float s = 0.0f, s2 = 0.0f;
#pragma unroll
    for (int v = 0; v < 8; ++v) {
        int row = rbase + v;
        float o = c[v] * nd[row] + bcol;
        outb[(size_t)row * 128 + col] = o;
        s  += o;
        s2 += o * o;
    }
    // fold rows 8-15 (lanes 16-31) into lanes 0-15, one atomic pair per column
    s  += __shfl_xor(s, 16, 32);
    s2 += __shfl_xor(s2, 16, 32);
    if (lane < 16) {
        atomicAdd(&stats[col], s);
        atomicAdd(&stats[128 + col], s2);
    }
}

// fold mean/var/gamma/beta into per-column scale & shift
__global__ void bn_prep_kernel(float* __restrict__ stats, const float* __restrict__ gamma,
                               const float* __restrict__ beta, float inv_n) {
    int c = threadIdx.x;  // 128 threads
    float mu  = stats[c] * inv_n;
    float var = stats[128 + c] * inv_n - mu * mu;
    float sc  = gamma[c] * rsqrtf(var + BN_EPS);
    stats[256 + c] = sc;
    stats[384 + c] = beta[c] - mu * sc;
}

// acc += out * scale + shift
__global__ void bn_acc_kernel(const float* __restrict__ outb, const float* __restrict__ stats,
                              float* __restrict__ acc, int total) {
    int i = blockIdx.x * blockDim.x + threadIdx.x;
    if (i < total) {
        int c = i & 127;
        acc[i] += outb[i] * stats[256 + c] + stats[384 + c];
    }
}

// out += resid
__global__ void resid_kernel(float* __restrict__ out, const float* __restrict__ feat, int total) {
    int i = blockIdx.x * blockDim.x + threadIdx.x;
    if (i < total) out[i] += feat[i];
}

// ---------------- host orchestration ----------------

extern "C" void kernel_launch(void* const* d_in, const int* in_sizes, int n_in,
                              void* d_out, int out_size, void* d_ws, size_t ws_size,
                              hipStream_t stream) {
    (void)in_sizes; (void)n_in; (void)out_size; (void)ws_size;

    const float* feat_in[3] = {(const float*)d_in[0], (const float*)d_in[1], (const float*)d_in[2]};
    const float* W     = (const float*)d_in[3];   // [2,9,128,128]
    const float* bias  = (const float*)d_in[4];   // [2,9,128]
    const float* gamma = (const float*)d_in[5];
    const float* beta  = (const float*)d_in[6];

    const int* esrc[9];
    const int* edst[9];
    for (int r = 0; r < 9; ++r) {
        esrc[r] = (const int*)d_in[7 + 2 * r];
        edst[r] = (const int*)d_in[8 + 2 * r];
    }

    static const int Nt[3] = {100000, 150000, 2000};                 // a, b, g
    static const int Ec[9] = {300000, 300000, 100000, 100000,
                              150000, 150000, 100000, 150000, 2000};
    static const int rs[9] = {0, 1, 0, 2, 1, 2, 0, 1, 2};            // src type per rel
    static const int rd[9] = {1, 0, 2, 0, 2, 1, 0, 1, 2};            // dst type per rel

    const size_t toff[3] = {0, (size_t)100000 * 128, (size_t)250000 * 128};
    const size_t NODE_TOTAL = (size_t)252000 * 128;

    // workspace layout (floats)
    float* ws = (float*)d_ws;
    size_t off = 0;
    float* ns[9];
    float* nd[9];
    for (int r = 0; r < 9; ++r) { ns[r] = ws + off; off += (size_t)Nt[rs[r]]; }
    for (int r = 0; r < 9; ++r) { nd[r] = ws + off; off += (size_t)Nt[rd[r]]; }
    float* mbuf  = ws + off; off += (size_t)150000 * 128;
    float* outb  = ws + off; off += (size_t)150000 * 128;
    float* stats = ws + off; off += 512;
    float* buf0  = ws + off; off += NODE_TOTAL;   // layer-0 output node features

    // ---- degrees (layer independent): int counts in-place -> rsqrt norms ----
    for (int r = 0; r < 9; ++r) {
        hipMemsetAsync(ns[r], 0, (size_t)Nt[rs[r]] * sizeof(float), stream);
        hipMemsetAsync(nd[r], 0, (size_t)Nt[rd[r]] * sizeof(float), stream);
    }
    for (int r = 0; r < 9; ++r) {
        deg_kernel<<<(Ec[r] + 255) / 256, 256, 0, stream>>>(
            esrc[r], edst[r], (int*)ns[r], (int*)nd[r], Ec[r]);
    }
    for (int r = 0; r < 9; ++r) {
        deg2norm_kernel<<<(Nt[rs[r]] + 255) / 256, 256, 0, stream>>>(ns[r], Nt[rs[r]]);
        deg2norm_kernel<<<(Nt[rd[r]] + 255) / 256, 256, 0, stream>>>(nd[r], Nt[rd[r]]);
    }

    float* outp = (float*)d_out;

    for (int l = 0; l < 2; ++l) {
        float* acc = (l == 0) ? buf0 : outp;     // layer 1 accumulates straight into d_out
        hipMemsetAsync(acc, 0, NODE_TOTAL * sizeof(float), stream);

        for (int r = 0; r < 9; ++r) {
            const int st = rs[r], dt = rd[r];
            const int ndst = Nt[dt];
            const int E = Ec[r];
            const float* fsrc = (l == 0) ? feat_in[st] : (buf0 + toff[st]);

            // m = segment_sum(feat[src] * ns[src], dst)
            hipMemsetAsync(mbuf, 0, (size_t)ndst * 128 * sizeof(float), stream);
            scatter_kernel<<<(E + 7) / 8, 256, 0, stream>>>(fsrc, ns[r], esrc[r], edst[r], mbuf, E);

            const size_t pidx = (size_t)(l * 9 + r);
            const float* Wr  = W     + pidx * 128 * 128;
            const float* br  = bias  + pidx * 128;
            const float* gr  = gamma + pidx * 128;
            const float* ber = beta  + pidx * 128;

            // out = (m @ W) * nd + bias, with BN sums fused into the epilogue
            hipMemsetAsync(stats, 0, 512 * sizeof(float), stream);
            gemm_bias_stats_kernel<<<ndst / 16, 256, 0, stream>>>(mbuf, Wr, nd[r], br, outb, stats);

            // finish batch-norm, accumulate into dst-type sum
            bn_prep_kernel<<<1, 128, 0, stream>>>(stats, gr, ber, 1.0f / (float)ndst);
            const int total = ndst * 128;
            bn_acc_kernel<<<(total + 255) / 256, 256, 0, stream>>>(outb, stats, acc + toff[dt], total);
        }
    }

    // residual: out += original features
    for (int t = 0; t < 3; ++t) {
        const int total = Nt[t] * 128;
        resid_kernel<<<(total + 255) / 256, 256, 0, stream>>>(outp + toff[t], feat_in[t], total);
    }
}